// ReuploadingRegressor_80925773791388
// MI455X (gfx1250) — compile-verified
//
#include <hip/hip_runtime.h>

// =====================================================================
// 10-qubit / 6-layer re-uploading circuit, batch 8192, on MI455X.
// Design: one wave32 per sample, state (1024 c64 = 8KB) resident in LDS.
//  - Per layer, fuse Rot*RY*RX into one 2x2 per wire (gates on distinct
//    wires commute), then Kronecker wires {0-3} and {4-7} into 16x16
//    unitaries applied as complex matmuls via V_WMMA_F32_16X16X4_F32.
//  - Wires {8,9} -> 4x4 unitary via VALU.
//  - CNOT ring = fixed index permutation, precomputed per lane, applied
//    as one gather/scatter LDS pass (folded into measurement last layer).
//  - Measurement + linear head folded into one |amp|^2 weighted sum.
// 4 waves (128 thr) per block, 32KB LDS/block, 2048 blocks. No barriers
// needed: each wave owns its LDS slice; DS ops are in-order per wave.
// =====================================================================

typedef float v2f __attribute__((ext_vector_type(2)));
typedef float v8f __attribute__((ext_vector_type(8)));

#define NQ 10
#define NLAYERS 6
#define DIM 1024
#define WPB 4           // waves per block
#define BATCH 8192

struct C2 { float re, im; };

__device__ __forceinline__ C2 cmul(C2 a, C2 b) {
  C2 r;
  r.re = fmaf(a.re, b.re, -a.im * b.im);
  r.im = fmaf(a.re, b.im,  a.im * b.re);
  return r;
}
__device__ __forceinline__ C2 cadd(C2 a, C2 b) { C2 r{a.re + b.re, a.im + b.im}; return r; }

struct M2 { C2 a00, a01, a10, a11; };

__device__ __forceinline__ M2 mm2(M2 A, M2 B) {
  M2 R;
  R.a00 = cadd(cmul(A.a00, B.a00), cmul(A.a01, B.a10));
  R.a01 = cadd(cmul(A.a00, B.a01), cmul(A.a01, B.a11));
  R.a10 = cadd(cmul(A.a10, B.a00), cmul(A.a11, B.a10));
  R.a11 = cadd(cmul(A.a10, B.a01), cmul(A.a11, B.a11));
  return R;
}

__device__ __forceinline__ M2 rxM(float t) {
  float s, c; __sincosf(0.5f * t, &s, &c);
  M2 R; R.a00 = {c, 0.f}; R.a01 = {0.f, -s}; R.a10 = {0.f, -s}; R.a11 = {c, 0.f};
  return R;
}
__device__ __forceinline__ M2 ryM(float t) {
  float s, c; __sincosf(0.5f * t, &s, &c);
  M2 R; R.a00 = {c, 0.f}; R.a01 = {-s, 0.f}; R.a10 = {s, 0.f}; R.a11 = {c, 0.f};
  return R;
}
// Rot(phi,theta,omega) = RZ(omega) RY(theta) RZ(phi), PennyLane convention.
__device__ __forceinline__ M2 rotM(float phi, float th, float om) {
  float st, ct, sa, ca, sb, cb;
  __sincosf(0.5f * th, &st, &ct);
  __sincosf(0.5f * (phi + om), &sa, &ca);
  __sincosf(0.5f * (phi - om), &sb, &cb);
  M2 R;
  R.a00 = { ca * ct, -sa * ct };   // e^{-i(phi+om)/2} cos
  R.a01 = { -cb * st, -sb * st };  // -e^{+i(phi-om)/2} sin
  R.a10 = { cb * st, -sb * st };   // e^{-i(phi-om)/2} sin
  R.a11 = { ca * ct,  sa * ct };   // e^{+i(phi+om)/2} cos
  return R;
}

__device__ __forceinline__ C2 gElem(const M2& G, int r, int c) {
  return r ? (c ? G.a11 : G.a10) : (c ? G.a01 : G.a00);
}

// amplitude index for state-matrix element (row K, flat col) per group:
// group 0: row = bits 9..6 (wires 0-3), col = bits 5..0
// group 1: row = bits 5..2 (wires 4-7), col = (bits 9..6, bits 1..0)
__device__ __forceinline__ int ampIdx(int g, int K, int colFull) {
  if (g == 0) return (K << 6) | colFull;
  int p = colFull >> 2, r = colFull & 3;
  return (p << 6) | (K << 2) | r;
}

__device__ __forceinline__ v8f wmma4(float a0, float a1, float b0, float b1, v8f c) {
  v2f A; A[0] = a0; A[1] = a1;
  v2f B; B[0] = b0; B[1] = b1;
  // (neg_a, A, neg_b, B, c_mod, C, reuse_a, reuse_b)
  return __builtin_amdgcn_wmma_f32_16x16x4_f32(false, A, false, B, (short)0, c, false, false);
}

__global__ __launch_bounds__(128, 1) void qcircuit_kernel(
    const float* __restrict__ x, const float* __restrict__ weights,
    const float* __restrict__ postW, const float* __restrict__ postB,
    const float* __restrict__ readout, const float* __restrict__ bias,
    float* __restrict__ out) {

  __shared__ __align__(16) float sm[WPB * 2 * DIM];   // 32 KB: re|im per wave
  const int lane = threadIdx.x & 31;
  const int wid  = threadIdx.x >> 5;
  const int b    = blockIdx.x * WPB + wid;

  float* smre = sm + wid * (2 * DIM);
  float* smim = smre + DIM;

  // ---- init |0...0>
  #pragma unroll
  for (int k = 0; k < 8; ++k) {
    int base = lane * 32 + k * 4;
    *(float4*)(smre + base) = make_float4(0.f, 0.f, 0.f, 0.f);
    *(float4*)(smim + base) = make_float4(0.f, 0.f, 0.f, 0.f);
  }
  if (lane == 0) smre[0] = 1.0f;

  // ---- per-sample encoding matrices (layer-independent)
  const float x0 = x[2 * b], x1 = x[2 * b + 1];
  const M2 E0 = mm2(ryM(x1), rxM(x0));   // even wires: RY(x1)*RX(x0)
  const M2 E1 = mm2(ryM(x0), rxM(x1));   // odd wires

  // ---- CNOT-ring gather permutation (same for every layer/sample)
  // apply gates in reverse order: i = g_first(...g_last(j)...)
  int perm[32];
  #pragma unroll
  for (int k = 0; k < 32; ++k) {
    int i = (k << 5) | lane;
    #pragma unroll
    for (int w = NQ - 1; w >= 0; --w) {         // gate w: ctrl wire w -> tgt wire (w+1)%NQ
      int cb = (NQ - 1) - w;                    // wire w = bit 9-w
      int tb = (w < NQ - 1) ? (NQ - 2 - w) : (NQ - 1);
      i ^= ((i >> cb) & 1) << tb;
    }
    perm[k] = i;
  }

  const int nIdx = lane & 15;              // WMMA column index this lane owns
  const int koff = (lane >> 4) << 1;       // K offset for upper lane half (A/B layout)
  const int mAdd = (lane >> 4) << 3;       // D rows: upper half holds M+8

  #pragma unroll 1
  for (int l = 0; l < NLAYERS; ++l) {
    const float* wt = weights + l * (NQ * 3);

    // ===== groups {wires 0-3} and {wires 4-7}: 16x16 complex matmul via WMMA
    #pragma unroll
    for (int g = 0; g < 2; ++g) {
      const float* wp = wt + g * 12;
      M2 G0 = mm2(rotM(wp[0],  wp[1],  wp[2]),  E0);
      M2 G1 = mm2(rotM(wp[3],  wp[4],  wp[5]),  E1);
      M2 G2 = mm2(rotM(wp[6],  wp[7],  wp[8]),  E0);
      M2 G3 = mm2(rotM(wp[9],  wp[10], wp[11]), E1);

      // row of each 2x2 selected by this lane's output-row bits (M)
      const int M = nIdx;
      C2 q0r0 = ((M >> 3) & 1) ? G0.a10 : G0.a00;
      C2 q0r1 = ((M >> 3) & 1) ? G0.a11 : G0.a01;
      C2 q1r0 = ((M >> 2) & 1) ? G1.a10 : G1.a00;
      C2 q1r1 = ((M >> 2) & 1) ? G1.a11 : G1.a01;
      C2 q2r0 = ((M >> 1) & 1) ? G2.a10 : G2.a00;
      C2 q2r1 = ((M >> 1) & 1) ? G2.a11 : G2.a01;
      C2 q3r0 = (M & 1) ? G3.a10 : G3.a00;
      C2 q3r1 = (M & 1) ? G3.a11 : G3.a01;

      // A-matrix fragments: U[M][K] = G0[m3][k3]G1[m2][k2]G2[m1][k1]G3[m0][k0]
      float urf[8], uif[8];
      #pragma unroll
      for (int v = 0; v < 8; ++v) {
        int K = 4 * (v >> 1) + (v & 1) + koff;
        C2 f0 = ((K >> 3) & 1) ? q0r1 : q0r0;
        C2 f1 = ((K >> 2) & 1) ? q1r1 : q1r0;
        C2 f2 = ((K >> 1) & 1) ? q2r1 : q2r0;
        C2 f3 = (K & 1) ? q3r1 : q3r0;
        C2 e = cmul(cmul(f0, f1), cmul(f2, f3));
        urf[v] = e.re; uif[v] = e.im;
      }

      // 4 column tiles of 16 (column blocks are independent -> in-place OK)
      #pragma unroll
      for (int c = 0; c < 4; ++c) {
        const int col = c * 16 + nIdx;
        float sr[8], si[8];
        #pragma unroll
        for (int v = 0; v < 8; ++v) {
          int K = 4 * (v >> 1) + (v & 1) + koff;
          int a = ampIdx(g, K, col);
          sr[v] = smre[a]; si[v] = smim[a];
        }
        v8f aR1 = {}, aR2 = {}, aI = {};
        #pragma unroll
        for (int t = 0; t < 4; ++t) aR1 = wmma4(urf[2*t], urf[2*t+1], sr[2*t], sr[2*t+1], aR1); // Ur*Sr
        #pragma unroll
        for (int t = 0; t < 4; ++t) aR2 = wmma4(uif[2*t], uif[2*t+1], si[2*t], si[2*t+1], aR2); // Ui*Si
        #pragma unroll
        for (int t = 0; t < 4; ++t) aI  = wmma4(urf[2*t], urf[2*t+1], si[2*t], si[2*t+1], aI);  // Ur*Si
        #pragma unroll
        for (int t = 0; t < 4; ++t) aI  = wmma4(uif[2*t], uif[2*t+1], sr[2*t], sr[2*t+1], aI);  // +Ui*Sr
        #pragma unroll
        for (int v = 0; v < 8; ++v) {
          int a = ampIdx(g, v + mAdd, col);
          smre[a] = aR1[v] - aR2[v];
          smim[a] = aI[v];
        }
      }
    }

    // ===== wires {8,9}: 4x4 complex unitary on bits 1..0 (VALU)
    {
      const float* wp = wt + 24;
      M2 G8 = mm2(rotM(wp[0], wp[1], wp[2]), E0);
      M2 G9 = mm2(rotM(wp[3], wp[4], wp[5]), E1);
      C2 uc[4][4];
      #pragma unroll
      for (int i = 0; i < 4; ++i)
        #pragma unroll
        for (int j = 0; j < 4; ++j)
          uc[i][j] = cmul(gElem(G8, (i >> 1) & 1, (j >> 1) & 1),
                          gElem(G9, i & 1, j & 1));
      #pragma unroll
      for (int k = 0; k < 8; ++k) {
        int base = ((k << 5) | lane) << 2;
        float4 r4 = *(const float4*)(smre + base);
        float4 i4 = *(const float4*)(smim + base);
        C2 s0{r4.x, i4.x}, s1{r4.y, i4.y}, s2{r4.z, i4.z}, s3{r4.w, i4.w};
        C2 n[4];
        #pragma unroll
        for (int i = 0; i < 4; ++i) {
          C2 acc = cmul(uc[i][0], s0);
          acc = cadd(acc, cmul(uc[i][1], s1));
          acc = cadd(acc, cmul(uc[i][2], s2));
          acc = cadd(acc, cmul(uc[i][3], s3));
          n[i] = acc;
        }
        *(float4*)(smre + base) = make_float4(n[0].re, n[1].re, n[2].re, n[3].re);
        *(float4*)(smim + base) = make_float4(n[0].im, n[1].im, n[2].im, n[3].im);
      }
    }

    // ===== CNOT ring = one permutation pass (skip after last layer; folded
    // into measurement). Wave-private slice + in-order DS: all loads below
    // are issued (and executed) before any store.
    if (l < NLAYERS - 1) {
      float tr[32], ti[32];
      #pragma unroll
      for (int k = 0; k < 32; ++k) { tr[k] = smre[perm[k]]; ti[k] = smim[perm[k]]; }
      asm volatile("" ::: "memory");   // keep compiler from sinking loads past stores
      #pragma unroll
      for (int k = 0; k < 32; ++k) {
        int j = (k << 5) | lane;
        smre[j] = tr[k]; smim[j] = ti[k];
      }
    }
  }

  // ===== measurement + linear head (final CNOT perm folded into gather)
  float W[NQ];
  #pragma unroll
  for (int w = 0; w < NQ; ++w) W[w] = postW[w];
  float totW = 0.f;
  #pragma unroll
  for (int w = 0; w < NQ; ++w) totW += W[w];

  float acc = 0.f;
  #pragma unroll
  for (int k = 0; k < 32; ++k) {
    int j = (k << 5) | lane;
    float rr = smre[perm[k]];
    float ii = smim[perm[k]];
    float p = fmaf(rr, rr, ii * ii);
    float msum = 0.f;
    #pragma unroll
    for (int w = 0; w < NQ; ++w)
      msum += ((j >> (NQ - 1 - w)) & 1) ? W[w] : 0.f;
    acc = fmaf(p, totW - 2.f * msum, acc);   // sum_w <Z_w> * W_w, folded
  }
  #pragma unroll
  for (int off = 16; off > 0; off >>= 1) acc += __shfl_xor(acc, off, 32);

  if (lane == 0) {
    out[b] = readout[0] * (acc + postB[0]) + bias[0];
  }
}

extern "C" void kernel_launch(void* const* d_in, const int* in_sizes, int n_in,
                              void* d_out, int out_size, void* d_ws, size_t ws_size,
                              hipStream_t stream) {
  (void)in_sizes; (void)n_in; (void)out_size; (void)d_ws; (void)ws_size;
  const float* x   = (const float*)d_in[0];
  const float* wts = (const float*)d_in[1];
  const float* pW  = (const float*)d_in[2];
  const float* pb  = (const float*)d_in[3];
  const float* ro  = (const float*)d_in[4];
  const float* bi  = (const float*)d_in[5];
  float* out = (float*)d_out;
  qcircuit_kernel<<<dim3(BATCH / WPB), dim3(32 * WPB), 0, stream>>>(x, wts, pW, pb, ro, bi, out);
}